// LinearAttention3D_70944269795771
// MI455X (gfx1250) — compile-verified
//
#include <hip/hip_runtime.h>

// ---------------------------------------------------------------------------
// Problem constants (fixed by the reference setup)
// ---------------------------------------------------------------------------
constexpr int B_SZ  = 2;
constexpr int C_SZ  = 64;
constexpr int NTOT  = 128 * 128 * 16;     // 262144 voxels per batch (2^18)
constexpr int NB    = 256;                // blocks per batch in reduce kernel
constexpr int WAVES_PER_BLOCK = 8;        // 256 threads, wave32
constexpr int TILES           = NTOT / 16;             // 16384 16-voxel tiles
constexpr int WAVES_PER_BATCH = NB * WAVES_PER_BLOCK;  // 2048
constexpr int TILES_PER_WAVE  = TILES / WAVES_PER_BATCH; // 8

typedef float v2f __attribute__((ext_vector_type(2)));
typedef float v4f __attribute__((ext_vector_type(4)));
typedef float v8f __attribute__((ext_vector_type(8)));

// merge two online-softmax partial triples {m, s, sv} (in place into a)
__device__ inline void merge3(float* a, const float* b) {
    float m  = fmaxf(a[0], b[0]);
    float ea = __expf(a[0] - m);
    float eb = __expf(b[0] - m);
    a[1] = a[1] * ea + b[1] * eb;
    a[2] = a[2] * ea + b[2] * eb;
    a[0] = m;
}

// ---------------------------------------------------------------------------
// Kernel 1: single pass over x. WMMA projection [4x64]x[64x16] per tile,
// online softmax accumulation, block-level reduce -> per-block partials.
// partial layout: [b][block][h][3] floats  (m, s, sv)
// ---------------------------------------------------------------------------
__global__ void __launch_bounds__(256)
la3d_reduce(const float* __restrict__ x,
            const float* __restrict__ Wqkv,
            float* __restrict__ partial)
{
    const int b        = blockIdx.y;
    const int lane     = threadIdx.x & 31;
    const int wave     = threadIdx.x >> 5;
    const int gw       = blockIdx.x * WAVES_PER_BLOCK + wave;
    const int laneN    = lane & 15;     // voxel within tile / M row
    const int laneHalf = lane >> 4;     // K sub-group select

    const float* xb = x + (size_t)b * C_SZ * NTOT;

    // Preload A fragments: A[m][k] = W_qkv[2+m][4*s + k] for m<4, zero otherwise.
    // Lane L, VGPR j holds A[L%16][4*s + j + 2*(L/16)].
    v2f a[16];
#pragma unroll
    for (int s = 0; s < 16; ++s) {
        float a0 = 0.0f, a1 = 0.0f;
        if (laneN < 4) {
            const float* wr = Wqkv + (size_t)(2 + laneN) * C_SZ + s * 4 + laneHalf * 2;
            a0 = wr[0];
            a1 = wr[1];
        }
        a[s][0] = a0;
        a[s][1] = a1;
    }

    // online softmax state per lane (lanes 0-15 carry real data)
    float m0 = -__builtin_inff(), s0 = 0.0f, sv0 = 0.0f;
    float m1 = -__builtin_inff(), s1 = 0.0f, sv1 = 0.0f;

    for (int t = 0; t < TILES_PER_WAVE; ++t) {
        const int tile = gw + t * WAVES_PER_BATCH;
        const int n0   = tile * 16;
        // B fragment base: lane L, VGPR j reads x[b][4*s + j + 2*laneHalf][n0 + laneN]
        const float* xp = xb + (size_t)(laneHalf * 2) * NTOT + n0 + laneN;

        v8f acc = {0.f, 0.f, 0.f, 0.f, 0.f, 0.f, 0.f, 0.f};
#pragma unroll
        for (int s = 0; s < 16; ++s) {
            v2f bf;
            bf[0] = __builtin_nontemporal_load(xp + (size_t)(4 * s + 0) * NTOT);
            bf[1] = __builtin_nontemporal_load(xp + (size_t)(4 * s + 1) * NTOT);
            acc = __builtin_amdgcn_wmma_f32_16x16x4_f32(
                false, a[s], false, bf, (short)0, acc, false, false);
        }

        // Lanes 0-15 (rows 0..3 of D): k_h0, k_h1, v_h0, v_h1 for voxel n0+laneN.
        // Lanes 16-31 see rows 8..11 which are exactly zero (A rows >=4 are 0);
        // they run the update harmlessly and are excluded from the reduction.
        const float k0  = acc[0];
        const float k1  = acc[1];
        const float vv0 = acc[2];
        const float vv1 = acc[3];

        float mn  = fmaxf(m0, k0);
        float eo  = __expf(m0 - mn);
        float en  = __expf(k0 - mn);
        s0  = s0 * eo + en;
        sv0 = sv0 * eo + en * vv0;
        m0  = mn;

        mn  = fmaxf(m1, k1);
        eo  = __expf(m1 - mn);
        en  = __expf(k1 - mn);
        s1  = s1 * eo + en;
        sv1 = sv1 * eo + en * vv1;
        m1  = mn;
    }

    // Block reduction over the 128 valid (wave, lane<16) entries.
    __shared__ float red[128 * 6];
    if (lane < 16) {
        float* r = &red[(wave * 16 + laneN) * 6];
        r[0] = m0; r[1] = s0; r[2] = sv0;
        r[3] = m1; r[4] = s1; r[5] = sv1;
    }
    __syncthreads();
#pragma unroll
    for (int stride = 64; stride >= 1; stride >>= 1) {
        if ((int)threadIdx.x < stride) {
            merge3(&red[threadIdx.x * 6 + 0], &red[(threadIdx.x + stride) * 6 + 0]);
            merge3(&red[threadIdx.x * 6 + 3], &red[(threadIdx.x + stride) * 6 + 3]);
        }
        __syncthreads();
    }
    if (threadIdx.x == 0) {
        float* p = partial + ((size_t)b * NB + blockIdx.x) * 6;
#pragma unroll
        for (int i = 0; i < 6; ++i) p[i] = red[i];
    }
}

// ---------------------------------------------------------------------------
// Kernel 2: merge per-block partials -> ctx[b][h] = sv/s, then compute the
// 128 output constants: cst[b*64+o] = Wout[o][0]*ctx[b][0]+Wout[o][1]*ctx[b][1]+bout[o]
// ---------------------------------------------------------------------------
__global__ void __launch_bounds__(128)
la3d_finalize(const float* __restrict__ partial,
              const float* __restrict__ Wout,
              const float* __restrict__ bout,
              float* __restrict__ cst)
{
    __shared__ float ctx[4];
    const int tid = threadIdx.x;
    if (tid < 4) {
        const int b = tid >> 1, h = tid & 1;
        float acc[3] = { -__builtin_inff(), 0.0f, 0.0f };
        for (int i = 0; i < NB; ++i) {
            const float* p = partial + ((size_t)b * NB + i) * 6 + h * 3;
            merge3(acc, p);
        }
        ctx[b * 2 + h] = acc[2] / acc[1];
    }
    __syncthreads();
    {
        const int b = tid >> 6, o = tid & 63;
        cst[b * 64 + o] = Wout[o * 2 + 0] * ctx[b * 2 + 0]
                        + Wout[o * 2 + 1] * ctx[b * 2 + 1]
                        + bout[o];
    }
}

// ---------------------------------------------------------------------------
// Kernel 3: broadcast the per-(b, channel) constant over all voxels.
// 134 MB of nontemporal float4 stores.
// ---------------------------------------------------------------------------
__global__ void __launch_bounds__(256)
la3d_broadcast(const float* __restrict__ cst, float* __restrict__ out)
{
    const size_t idx = (size_t)blockIdx.x * blockDim.x + threadIdx.x;
    const size_t e   = idx * 4;
    const int    row = (int)(e >> 18);           // NTOT == 2^18
    const float  v   = cst[row];
    v4f val = { v, v, v, v };
    __builtin_nontemporal_store(val, reinterpret_cast<v4f*>(out) + idx);
}

// ---------------------------------------------------------------------------
// Host launcher
// ---------------------------------------------------------------------------
extern "C" void kernel_launch(void* const* d_in, const int* in_sizes, int n_in,
                              void* d_out, int out_size, void* d_ws, size_t ws_size,
                              hipStream_t stream)
{
    const float* x     = (const float*)d_in[0];
    // d_in[1] = query (unused: softmax over dim_head==1 makes q identically 1)
    const float* Wqkv  = (const float*)d_in[2];
    // d_in[3] = W_q (unused for the same reason)
    const float* Wout  = (const float*)d_in[4];
    const float* bout  = (const float*)d_in[5];
    float*       out   = (float*)d_out;

    float* partial = (float*)d_ws;                 // B_SZ*NB*6 = 3072 floats
    float* cst     = partial + (size_t)B_SZ * NB * 6; // 128 floats

    dim3 grid1(NB, B_SZ);
    la3d_reduce<<<grid1, 256, 0, stream>>>(x, Wqkv, partial);
    la3d_finalize<<<1, 128, 0, stream>>>(partial, Wout, bout, cst);

    const int total4 = (B_SZ * C_SZ * NTOT) / 4;   // 8388608
    la3d_broadcast<<<total4 / 256, 256, 0, stream>>>(cst, out);
}